// GeneralNet_38878043963420
// MI455X (gfx1250) — compile-verified
//
#include <hip/hip_runtime.h>
#include <hip/hip_bf16.h>

typedef __attribute__((ext_vector_type(2))) float v2f;
typedef __attribute__((ext_vector_type(8))) float v8f;

#define DFEAT 128

// ---------------------------------------------------------------------------
// Zero a contiguous float region (grid-stride).
__global__ __launch_bounds__(256)
void zero_kernel(float* __restrict__ p, size_t n) {
    size_t i = (size_t)blockIdx.x * blockDim.x + threadIdx.x;
    size_t stride = (size_t)gridDim.x * blockDim.x;
    for (; i < n; i += stride) p[i] = 0.0f;
}

// ---------------------------------------------------------------------------
// Edge scatter-add: one wave32 per edge. Lane j gathers x[src][4j..4j+3] as a
// float4 (512B fully-coalesced per wave) and atomically adds into agg[dst].
// Lane 0 bumps the degree counter. ei is [2, E] int64 row-major.
__global__ __launch_bounds__(256)
void scatter_kernel(const float* __restrict__ x, const long long* __restrict__ ei,
                    float* __restrict__ agg, float* __restrict__ cnt, int E) {
    int e = blockIdx.x * 8 + (threadIdx.x >> 5);
    if (e >= E) return;
    int lane = threadIdx.x & 31;
    long long s = ei[e];
    long long d = ei[(size_t)E + e];
    const float4 v = *(const float4*)(x + s * DFEAT + lane * 4);
    float* dp = agg + d * DFEAT + lane * 4;
    atomicAdd(dp + 0, v.x);
    atomicAdd(dp + 1, v.y);
    atomicAdd(dp + 2, v.z);
    atomicAdd(dp + 3, v.w);
    if (lane == 0) atomicAdd(cnt + d, 1.0f);
}

// ---------------------------------------------------------------------------
// agg[i] /= max(cnt[row], 1)
__global__ __launch_bounds__(256)
void normalize_kernel(float* __restrict__ agg, const float* __restrict__ cnt, int n) {
    int i = blockIdx.x * blockDim.x + threadIdx.x;
    if (i >= n) return;
    float c = cnt[i >> 7];   // row = i / 128
    agg[i] *= 1.0f / fmaxf(c, 1.0f);
}

// ---------------------------------------------------------------------------
// K-loop of V_WMMA_F32_16X16X4_F32 accumulating A(16xK) x W^T(Kx16) into c.
// Arow/Wrow already include the lane's +2*(lane>>4) k-phase offset.
__device__ __forceinline__
v8f wmma_accum_row(v8f c, const float* __restrict__ Arow,
                   const float* __restrict__ Wrow, int K) {
#pragma unroll 4
    for (int k = 0; k < K; k += 4) {
        v2f a = *(const v2f*)(Arow + k);   // A[m][k+2hi], A[m][k+2hi+1]
        v2f b = *(const v2f*)(Wrow + k);   // W[n][k+2hi], W[n][k+2hi+1] == B[k][n]
        c = __builtin_amdgcn_wmma_f32_16x16x4_f32(
                /*neg_a=*/false, a, /*neg_b=*/false, b,
                /*c_mod=*/(short)0, c, /*reuse_a=*/false, /*reuse_b=*/false);
    }
    return c;
}

// ---------------------------------------------------------------------------
// Fused multi-term linear: out[N,OC] = sum_t A_t[N,K] @ W_t[OC,K]^T
//                                      (+ bias0[OC] + bias1[OC]) (opt. ReLU)
// grid.x = row tiles (16 rows), grid.y = 128-col groups. 8 waves/block; wave w
// owns 16 output columns. All branching on nterms/bias is wave-uniform, so
// EXEC is all-ones around every WMMA as the ISA requires.
__global__ __launch_bounds__(256)
void fused_linear_wmma(const float* __restrict__ A0, const float* __restrict__ W0,
                       const float* __restrict__ A1, const float* __restrict__ W1,
                       const float* __restrict__ A2, const float* __restrict__ W2,
                       const float* __restrict__ bias0, const float* __restrict__ bias1,
                       float* __restrict__ out,
                       int N, int K, int OC, int nterms, int do_relu) {
    const int lane  = threadIdx.x & 31;
    const int wave  = threadIdx.x >> 5;
    const int mbase = blockIdx.x * 16;
    const int nbase = blockIdx.y * 128 + wave * 16;
    if (nbase >= OC) return;                 // wave-uniform

    const int hi = lane >> 4;                // k-phase: 0 -> K{0,1}, 1 -> K{2,3}
    const int lo = lane & 15;
    int arow = mbase + lo; if (arow >= N) arow = N - 1;   // keep EXEC full
    const int wrow = nbase + lo;
    const size_t aoff = (size_t)arow * K + 2 * hi;
    const size_t woff = (size_t)wrow * K + 2 * hi;

    v8f c = {0.f, 0.f, 0.f, 0.f, 0.f, 0.f, 0.f, 0.f};
    c = wmma_accum_row(c, A0 + aoff, W0 + woff, K);
    if (nterms > 1) c = wmma_accum_row(c, A1 + aoff, W1 + woff, K);
    if (nterms > 2) c = wmma_accum_row(c, A2 + aoff, W2 + woff, K);

    float bv = 0.0f;
    if (bias0) bv += bias0[nbase + lo];
    if (bias1) bv += bias1[nbase + lo];

#pragma unroll
    for (int r = 0; r < 8; ++r) {
        int row = mbase + r + 8 * hi;        // C layout: VGPR r, halves of wave
        if (row < N) {
            float v = c[r] + bv;
            if (do_relu) v = fmaxf(v, 0.0f);
            out[(size_t)row * OC + nbase + lo] = v;
        }
    }
}

// ---------------------------------------------------------------------------
// Head: out[n,0:3] = h[n,0:256] @ W2[3,256]^T + b2. One wave per node,
// __shfl_xor tree reduction across 32 lanes.
__global__ __launch_bounds__(256)
void head_kernel(const float* __restrict__ h, const float* __restrict__ W2,
                 const float* __restrict__ b2, float* __restrict__ out, int N) {
    int node = blockIdx.x * 8 + (threadIdx.x >> 5);
    if (node >= N) return;
    int lane = threadIdx.x & 31;
    const float* hrow = h + (size_t)node * 256;
    float hv[8];
#pragma unroll
    for (int j = 0; j < 8; ++j) hv[j] = hrow[lane + 32 * j];
#pragma unroll
    for (int o = 0; o < 3; ++o) {
        const float* w = W2 + o * 256;
        float p = 0.0f;
#pragma unroll
        for (int j = 0; j < 8; ++j) p += hv[j] * w[lane + 32 * j];
#pragma unroll
        for (int off = 16; off > 0; off >>= 1) p += __shfl_xor(p, off, 32);
        if (lane == 0) out[(size_t)node * 3 + o] = p + b2[o];
    }
}

// ---------------------------------------------------------------------------
extern "C" void kernel_launch(void* const* d_in, const int* in_sizes, int n_in,
                              void* d_out, int out_size, void* d_ws, size_t ws_size,
                              hipStream_t stream) {
    const float*     x   = (const float*)d_in[0];
    const long long* ei0 = (const long long*)d_in[1];
    const long long* ei1 = (const long long*)d_in[2];
    const float* Wl1 = (const float*)d_in[3];
    const float* bl1 = (const float*)d_in[4];
    const float* Wr1 = (const float*)d_in[5];
    const float* Wl2 = (const float*)d_in[6];
    const float* bl2 = (const float*)d_in[7];
    const float* Wr2 = (const float*)d_in[8];
    const float* Wsk = (const float*)d_in[9];
    const float* bsk = (const float*)d_in[10];
    const float* W1  = (const float*)d_in[11];
    const float* b1  = (const float*)d_in[12];
    const float* W2  = (const float*)d_in[13];
    const float* b2  = (const float*)d_in[14];

    const int N = in_sizes[0] / DFEAT;   // 50000
    const int E = in_sizes[1] / 2;       // 800000
    const int H = 256;

    // Workspace layout (floats)
    float* AGG = (float*)d_ws;                       // N*128
    float* CNT = AGG + (size_t)N * DFEAT;            // N   (contiguous with AGG)
    float* H1  = CNT + N;                            // N*128
    float* H3  = H1  + (size_t)N * DFEAT;            // N*128
    float* H4  = H3  + (size_t)N * DFEAT;            // N*256
    float* OUT = (float*)d_out;

    const int rowTiles   = (N + 15) / 16;            // 3125
    const int edgeBlocks = (E + 7) / 8;              // 100000
    const int ndElems    = N * DFEAT;
    const int ndBlocks   = (ndElems + 255) / 256;

    // ---- Layer 1: mean-aggregate then h1 = agg@Wl1^T + x@Wr1^T + bl1 ----
    zero_kernel<<<2048, 256, 0, stream>>>(AGG, (size_t)N * (DFEAT + 1));
    scatter_kernel<<<edgeBlocks, 256, 0, stream>>>(x, ei0, AGG, CNT, E);
    normalize_kernel<<<ndBlocks, 256, 0, stream>>>(AGG, CNT, ndElems);
    fused_linear_wmma<<<dim3(rowTiles, 1), 256, 0, stream>>>(
        AGG, Wl1, x, Wr1, nullptr, nullptr, bl1, nullptr,
        H1, N, DFEAT, DFEAT, /*nterms=*/2, /*relu=*/0);

    // ---- Layer 2 + skip: h3 = agg2@Wl2^T + h1@Wr2^T + x@Ws^T + bl2 + bs ----
    zero_kernel<<<2048, 256, 0, stream>>>(AGG, (size_t)N * (DFEAT + 1));
    scatter_kernel<<<edgeBlocks, 256, 0, stream>>>(H1, ei1, AGG, CNT, E);
    normalize_kernel<<<ndBlocks, 256, 0, stream>>>(AGG, CNT, ndElems);
    fused_linear_wmma<<<dim3(rowTiles, 1), 256, 0, stream>>>(
        AGG, Wl2, H1, Wr2, x, Wsk, bl2, bsk,
        H3, N, DFEAT, DFEAT, /*nterms=*/3, /*relu=*/0);

    // ---- MLP hidden: h4 = relu(h3 @ W1^T + b1), OC = 256 ----
    fused_linear_wmma<<<dim3(rowTiles, 2), 256, 0, stream>>>(
        H3, W1, nullptr, nullptr, nullptr, nullptr, b1, nullptr,
        H4, N, DFEAT, H, /*nterms=*/1, /*relu=*/1);

    // ---- Head: out = h4 @ W2^T + b2 ----
    head_kernel<<<(N + 7) / 8, 256, 0, stream>>>(H4, W2, b2, OUT, N);

    (void)n_in; (void)out_size; (void)ws_size;
}